// ContrastiveLoss_36704790512372
// MI455X (gfx1250) — compile-verified
//
#include <hip/hip_runtime.h>
#include <hip/hip_bf16.h>
#include <math.h>

// Problem constants (match reference)
#define B_N     4096
#define D_DIM   1024
#define S_NUM   5
#define MARGIN  1.0f
#define EPS_F   1e-6f
#define NTILES  (B_N / 16)   // 256 row tiles

typedef __attribute__((ext_vector_type(2))) float v2f;
typedef __attribute__((ext_vector_type(8))) float v8f;

// ---------------------------------------------------------------------------
// Kernel 1: per-row stats.  stats layout in d_ws (floats):
//   [0,B)    vnorm  = ||v_k||^2
//   [B,2B)   anorm  = ||a_k||^2
//   [2B,3B)  vsum   = sum(v_k)
//   [3B,4B)  asum   = sum(a_k)
// One wave per row (8192 rows total), coalesced float4 loads.
// ---------------------------------------------------------------------------
__global__ __launch_bounds__(256)
void contrastive_row_stats(const float* __restrict__ vfeat,
                           const float* __restrict__ afeat,
                           float* __restrict__ stats) {
    const int gtid = blockIdx.x * blockDim.x + threadIdx.x;
    const int row  = gtid >> 5;          // global wave id = row (0..8191)
    const int lane = threadIdx.x & 31;
    if (row >= 2 * B_N) return;

    const float* src = (row < B_N)
        ? (vfeat + (size_t)row * D_DIM)
        : (afeat + (size_t)(row - B_N) * D_DIM);

    float n2 = 0.0f, sm = 0.0f;
    // lanes cover 128 consecutive floats per sweep (32 lanes x float4), 8 sweeps
    for (int i = lane * 4; i < D_DIM; i += 32 * 4) {
        float4 x = *(const float4*)(src + i);
        n2 += x.x * x.x + x.y * x.y + x.z * x.z + x.w * x.w;
        sm += x.x + x.y + x.z + x.w;
    }
    #pragma unroll
    for (int sh = 16; sh >= 1; sh >>= 1) {
        n2 += __shfl_xor(n2, sh, 32);
        sm += __shfl_xor(sm, sh, 32);
    }
    if (lane == 0) {
        if (row < B_N) {
            stats[row]             = n2;   // vnorm
            stats[2 * B_N + row]   = sm;   // vsum
        } else {
            const int r = row - B_N;
            stats[B_N + r]         = n2;   // anorm
            stats[3 * B_N + r]     = sm;   // asum
        }
    }
}

// ---------------------------------------------------------------------------
// Kernel 2: banded Gram via V_WMMA_F32_16X16X4_F32 + hinge epilogue.
// One workgroup per 16-row tile t; 4 waves split K=1024 into 256-wide chunks.
// Per wave, per k-step (K=4):
//   a0 = V[rows t]      fragment   (16x4)
//   a1 = V[rows t+1]    fragment
//   b0 = A[cols t]^T    fragment   (4x16)
//   bp = A[cols t+1]^T  fragment
//   c0 += a0*b0 ; cp += a0*bp ; cn += a1*b0
// Partial C tiles combined in LDS with ds_add_f32; lanes 0..15 compute the
// per-anchor hinge from the Gram band + row stats; one atomicAdd per WG.
//
// Fragment layout assumptions (CDNA5 ISA 7.12.2):
//   f32 A 16x4 : lane L holds (M = L&15, K = 2*(L>>4)+v), v=0,1  -> float2 load
//   f32 B 4x16 : lane L holds (N = L&15, K = 2*(L>>4)+v)         -> float2 load
//   f32 C 16x16: lane L, vgpr v holds (M = v + 8*(L>>4), N = L&15)
// ---------------------------------------------------------------------------
__global__ __launch_bounds__(128)
void contrastive_band_hinge(const float* __restrict__ vfeat,
                            const float* __restrict__ afeat,
                            const float* __restrict__ stats,
                            float* __restrict__ out) {
    __shared__ float T0[16][16];   // G[rows t,   cols t  ]
    __shared__ float T1[16][16];   // G[rows t,   cols t+1]
    __shared__ float T3[16][16];   // G[rows t+1, cols t  ]

    const int t    = blockIdx.x;          // row tile
    const int tid  = threadIdx.x;
    const int wave = tid >> 5;            // 0..3 (K-split)
    const int lane = tid & 31;
    const int half = lane >> 4;           // 0/1
    const int l15  = lane & 15;

    const int row0 = t * 16;
    const int rowN = (row0 + 16) % B_N;   // rows of tile t+1 (wraps)

    // zero LDS accumulators
    for (int i = tid; i < 256; i += 128) {
        (&T0[0][0])[i] = 0.0f;
        (&T1[0][0])[i] = 0.0f;
        (&T3[0][0])[i] = 0.0f;
    }
    __syncthreads();

    // per-lane fragment base pointers (A and B share the same lane pattern)
    const float* pa0 = vfeat + (size_t)(row0 + l15) * D_DIM + half * 2;
    const float* pa1 = vfeat + (size_t)(rowN + l15) * D_DIM + half * 2;
    const float* pb0 = afeat + (size_t)(row0 + l15) * D_DIM + half * 2;
    const float* pbp = afeat + (size_t)(rowN + l15) * D_DIM + half * 2;

    v8f c0 = {};
    v8f cp = {};
    v8f cn = {};

    const int kbeg = wave * (D_DIM / 4);          // 256-wide K chunk per wave
    const int kend = kbeg + (D_DIM / 4);
    for (int kk = kbeg; kk < kend; kk += 4) {
        v2f a0 = *(const v2f*)(pa0 + kk);
        v2f a1 = *(const v2f*)(pa1 + kk);
        v2f b0 = *(const v2f*)(pb0 + kk);
        v2f bp = *(const v2f*)(pbp + kk);
        c0 = __builtin_amdgcn_wmma_f32_16x16x4_f32(false, a0, false, b0,
                                                   (short)0, c0, false, false);
        cp = __builtin_amdgcn_wmma_f32_16x16x4_f32(false, a0, false, bp,
                                                   (short)0, cp, false, false);
        cn = __builtin_amdgcn_wmma_f32_16x16x4_f32(false, a1, false, b0,
                                                   (short)0, cn, false, false);
    }

    // combine partial C tiles across the 4 waves in LDS (ds_add_f32)
    #pragma unroll
    for (int v = 0; v < 8; ++v) {
        const int M = v + 8 * half;
        atomicAdd(&T0[M][l15], c0[v]);
        atomicAdd(&T1[M][l15], cp[v]);
        atomicAdd(&T3[M][l15], cn[v]);
    }
    __syncthreads();

    // ---- hinge epilogue: lanes 0..15 of wave 0, one anchor each ----
    if (wave == 0) {
        float part = 0.0f;
        if (lane < 16) {
            const float* vnorm = stats;
            const float* anorm = stats + B_N;
            const float* vsum  = stats + 2 * B_N;
            const float* asum  = stats + 3 * B_N;
            const float  ceps  = (float)D_DIM * EPS_F * EPS_F;

            const int k = row0 + lane;

            const float Gkk = T0[lane][lane];
            float dp2 = vnorm[k] + anorm[k] - 2.0f * Gkk
                      + 2.0f * EPS_F * (vsum[k] - asum[k]) + ceps;
            const float d_p = sqrtf(fmaxf(dp2, 0.0f));

            float dn1 = 3.0e38f, dn2 = 3.0e38f;
            #pragma unroll
            for (int m = 0; m < S_NUM; ++m) {
                const int off = (m == k) ? 1 : (m + 1);       // 1..5
                const int j   = (k + off) % B_N;
                const int cl  = lane + off;                   // 1..20
                const float Gkj = (cl < 16) ? T0[lane][cl] : T1[lane][cl - 16];
                const float Gjk = (cl < 16) ? T0[cl][lane] : T3[cl - 16][lane];

                float a2 = vnorm[k] + anorm[j] - 2.0f * Gkj
                         + 2.0f * EPS_F * (vsum[k] - asum[j]) + ceps;
                float b2 = anorm[k] + vnorm[j] - 2.0f * Gjk
                         + 2.0f * EPS_F * (asum[k] - vsum[j]) + ceps;
                dn1 = fminf(dn1, sqrtf(fmaxf(a2, 0.0f)));
                dn2 = fminf(dn2, sqrtf(fmaxf(b2, 0.0f)));
            }
            const float hinge = fmaxf(MARGIN + 2.0f * d_p - dn1 - dn2, 0.0f);
            part = hinge * (1.0f / (float)B_N);
        }
        #pragma unroll
        for (int sh = 16; sh >= 1; sh >>= 1)
            part += __shfl_xor(part, sh, 32);
        if (lane == 0)
            atomicAdd(out, part);
    }
}

// ---------------------------------------------------------------------------
extern "C" void kernel_launch(void* const* d_in, const int* in_sizes, int n_in,
                              void* d_out, int out_size, void* d_ws, size_t ws_size,
                              hipStream_t stream) {
    (void)in_sizes; (void)n_in; (void)out_size; (void)ws_size;

    const float* vfeat = (const float*)d_in[0];
    const float* afeat = (const float*)d_in[1];
    float*       out   = (float*)d_out;
    float*       stats = (float*)d_ws;          // 4 * B_N floats = 64 KB

    // output is accumulated via atomics -> must start at zero every call
    hipMemsetAsync(out, 0, sizeof(float), stream);

    // 8192 rows, 8 waves (256 threads) per block -> 1024 blocks
    contrastive_row_stats<<<(2 * B_N) / 8, 256, 0, stream>>>(vfeat, afeat, stats);

    // one workgroup (4 waves) per 16-row tile
    contrastive_band_hinge<<<NTILES, 128, 0, stream>>>(vfeat, afeat, stats, out);
}